// ModuleSoftsplat_283467842192
// MI455X (gfx1250) — compile-verified
//
#include <hip/hip_runtime.h>

// Problem geometry (fixed by the reference).
namespace {
constexpr int B = 4, C = 64, H = 256, W = 448;
constexpr int HW   = H * W;          // 114688
constexpr int NPIX = B * HW;         // 458752
constexpr long long OUT_N = (long long)B * C * HW;  // 29360128
}

// Zero the output accumulator and the norm plane (harness poisons them).
__global__ void zero_ws_kernel(float4* __restrict__ out4,
                               float4* __restrict__ norm4,
                               int nOut4, int nNorm4) {
    int tid    = blockIdx.x * blockDim.x + threadIdx.x;
    int stride = gridDim.x * blockDim.x;
    float4 z = {0.f, 0.f, 0.f, 0.f};
    for (int i = tid; i < nOut4;  i += stride) out4[i]  = z;
    for (int i = tid; i < nNorm4; i += stride) norm4[i] = z;
}

// One thread per source pixel: splat m to norm plane, input*m to 64 channels,
// 4 bilinear corners each, via native f32 global atomics (L2-resident).
__global__ void softsplat_scatter_kernel(const float* __restrict__ input,
                                         const float* __restrict__ flow,
                                         const float* __restrict__ metric,
                                         float* __restrict__ out,
                                         float* __restrict__ norm) {
    int pix = blockIdx.x * blockDim.x + threadIdx.x;
    if (pix >= NPIX) return;
    int b = pix / HW;
    int p = pix - b * HW;
    int y = p / W;
    int x = p - y * W;

    const float fx = flow[(size_t)b * 2 * HW + p];
    const float fy = flow[(size_t)b * 2 * HW + HW + p];
    const float m  = expf(metric[pix]);          // metric is (B,1,H,W) == pix-flat

    // Match the reference's exact weight formulation.
    float fltX = (float)x + fx;
    float fltY = (float)y + fy;
    float x0f = floorf(fltX), y0f = floorf(fltY);
    float x1f = x0f + 1.0f,   y1f = y0f + 1.0f;
    int x0  = (int)x0f, y0i = (int)y0f;
    int x1  = x0 + 1,   y1i = y0i + 1;
    float wx0 = x1f - fltX, wx1 = fltX - x0f;
    float wy0 = y1f - fltY, wy1 = fltY - y0f;

    const int   xs[4] = {x0,  x1,  x0,  x1};
    const int   ys[4] = {y0i, y0i, y1i, y1i};
    const float ws[4] = {wx0 * wy0, wx1 * wy0, wx0 * wy1, wx1 * wy1};

    int   idx[4];
    float wgt[4];
#pragma unroll
    for (int k = 0; k < 4; ++k) {
        bool valid = (xs[k] >= 0) && (xs[k] < W) && (ys[k] >= 0) && (ys[k] < H);
        idx[k] = valid ? (ys[k] * W + xs[k]) : -1;
        wgt[k] = ws[k];
    }

    // Norm-channel splat (once per pixel).
    float* nrm = norm + (size_t)b * HW;
#pragma unroll
    for (int k = 0; k < 4; ++k)
        if (idx[k] >= 0) unsafeAtomicAdd(nrm + idx[k], m * wgt[k]);

    // 64-channel splat: coalesced strided loads (stride HW across channels,
    // contiguous across lanes), gfx1250 global_prefetch to cover latency.
    const float* inp  = input + (size_t)b * C * HW + p;
    float*       outb = out   + (size_t)b * C * HW;
#pragma unroll 4
    for (int c = 0; c < C; ++c) {
        int pc = (c + 8 < C) ? c + 8 : C - 1;
        __builtin_prefetch(inp + (size_t)pc * HW, 0, 3);   // global_prefetch_b8
        float  v  = inp[(size_t)c * HW] * m;
        float* oc = outb + (size_t)c * HW;
        if (idx[0] >= 0) unsafeAtomicAdd(oc + idx[0], v * wgt[0]);
        if (idx[1] >= 0) unsafeAtomicAdd(oc + idx[1], v * wgt[1]);
        if (idx[2] >= 0) unsafeAtomicAdd(oc + idx[2], v * wgt[2]);
        if (idx[3] >= 0) unsafeAtomicAdd(oc + idx[3], v * wgt[3]);
    }
}

// out[b,c,y,x] /= (norm[b,y,x]==0 ? 1 : norm[b,y,x]); true division to match
// the reference's rounding.
__global__ void normalize_kernel(float* __restrict__ out,
                                 const float* __restrict__ norm) {
    int pix = blockIdx.x * blockDim.x + threadIdx.x;
    if (pix >= NPIX) return;
    int b = pix / HW;
    int p = pix - b * HW;
    float n = norm[pix];
    n = (n == 0.0f) ? 1.0f : n;
    float* outb = out + (size_t)b * C * HW + p;
#pragma unroll 8
    for (int c = 0; c < C; ++c)
        outb[(size_t)c * HW] /= n;
}

extern "C" void kernel_launch(void* const* d_in, const int* in_sizes, int n_in,
                              void* d_out, int out_size, void* d_ws, size_t ws_size,
                              hipStream_t stream) {
    const float* input  = (const float*)d_in[0];  // (4,64,256,448) f32
    const float* flow   = (const float*)d_in[1];  // (4,2,256,448)  f32
    const float* metric = (const float*)d_in[2];  // (4,1,256,448)  f32
    float* out  = (float*)d_out;                  // (4,64,256,448) f32
    float* norm = (float*)d_ws;                   // (4,256,448)    f32 scratch

    const int nOut4  = (int)(OUT_N / 4);
    const int nNorm4 = NPIX / 4;
    zero_ws_kernel<<<2048, 256, 0, stream>>>((float4*)out, (float4*)norm,
                                             nOut4, nNorm4);

    const int blocks = (NPIX + 255) / 256;        // 1792 blocks, 8 waves each
    softsplat_scatter_kernel<<<blocks, 256, 0, stream>>>(input, flow, metric,
                                                         out, norm);
    normalize_kernel<<<blocks, 256, 0, stream>>>(out, norm);
}